// Quantizer_12610023981628
// MI455X (gfx1250) — compile-verified
//
#include <hip/hip_runtime.h>
#include <hip/hip_bf16.h>

#define N_EMBED   1024
#define EMBED_DIM 256
#define HW        1024        // 32*32
#define M_ROWS    32768       // 32*HW
#define X_BSTRIDE 262144      // EMBED_DIM*HW
#define X_TOTAL   8388608     // 32*EMBED_DIM*HW
#define K_STEPS   64          // EMBED_DIM / 4

typedef __attribute__((ext_vector_type(2))) float v2f;
typedef __attribute__((ext_vector_type(8))) float v8f;

// ---------------- kernel 1: per-code squared norms ----------------
__global__ void code_norm_kernel(const float* __restrict__ weight,
                                 float* __restrict__ cn) {
    __shared__ float red[256];
    int e = blockIdx.x, t = threadIdx.x;
    float v = weight[e * EMBED_DIM + t];
    red[t] = v * v;
    __syncthreads();
    for (int s = 128; s > 0; s >>= 1) {
        if (t < s) red[t] += red[t + s];
        __syncthreads();
    }
    if (t == 0) cn[e] = red[0];
}

// ---------------- kernel 2: fused WMMA distance + argmin ----------------
// block = 256 threads = 8 waves; each wave owns a 16-row tile of flat and
// processes code tiles 4-at-a-time: per K-step one A fragment feeds 4 WMMAs
// against 4 B fragments (immediate-offset b64 loads). ~75 VGPRs, no spills.
__global__ void __launch_bounds__(256, 1)
argmin_wmma_kernel(const float* __restrict__ x,
                   const float* __restrict__ weight,
                   const float* __restrict__ cn,
                   float* __restrict__ idx_out_f,
                   int* __restrict__ idx_out_i) {
    int lane = threadIdx.x & 31;
    int wave = threadIdx.x >> 5;
    int r0   = blockIdx.x * 128 + wave * 16;   // first row of this wave's tile
    int m    = lane & 15;                      // A row-in-tile / B col-in-tile
    int hl   = lane >> 4;                      // half-wave select (K offset 0 or 2)

    // flat[row, k] == x[b, k, h, w]; within a 16-row tile b is constant and hw
    // is contiguous, so A-fragment loads coalesce across lanes 0-15 / 16-31.
    int row = r0 + m;
    int b   = row >> 10;
    int hw  = row & 1023;
    const float* xa = x + (size_t)b * X_BSTRIDE + hw + (size_t)(2 * hl) * HW;

    float best[8];
    int   bidx[8];
#pragma unroll
    for (int v = 0; v < 8; ++v) { best[v] = 3.4e38f; bidx[v] = 0; }

    for (int ng = 0; ng < N_EMBED; ng += 64) {        // group of 4 code tiles
        const float* wb = weight + (size_t)(ng + m) * EMBED_DIM + 2 * hl;
        if (ng + 64 < N_EMBED)
            __builtin_prefetch(wb + 64 * EMBED_DIM, 0, 1);

        v8f acc0 = {}, acc1 = {}, acc2 = {}, acc3 = {};
#pragma unroll 4
        for (int ks = 0; ks < K_STEPS; ++ks) {
            // A 16x4 fragment (ISA 7.12.2): lane m holds K = 4*ks+2*hl, +1
            v2f a;
            a.x = xa[(size_t)(4 * ks) * HW];
            a.y = xa[(size_t)(4 * ks + 1) * HW];
            // B 4x16 fragments for 4 code tiles: col = m, contiguous K pair;
            // tiles are 16*EMBED_DIM floats (16 KB) apart -> immediate offsets
            v2f b0, b1, b2, b3;
            b0.x = wb[4 * ks];
            b0.y = wb[4 * ks + 1];
            b1.x = wb[4 * ks + 16 * EMBED_DIM];
            b1.y = wb[4 * ks + 16 * EMBED_DIM + 1];
            b2.x = wb[4 * ks + 32 * EMBED_DIM];
            b2.y = wb[4 * ks + 32 * EMBED_DIM + 1];
            b3.x = wb[4 * ks + 48 * EMBED_DIM];
            b3.y = wb[4 * ks + 48 * EMBED_DIM + 1];
            acc0 = __builtin_amdgcn_wmma_f32_16x16x4_f32(
                false, a, false, b0, (short)0, acc0, false, false);
            acc1 = __builtin_amdgcn_wmma_f32_16x16x4_f32(
                false, a, false, b1, (short)0, acc1, false, false);
            acc2 = __builtin_amdgcn_wmma_f32_16x16x4_f32(
                false, a, false, b2, (short)0, acc2, false, false);
            acc3 = __builtin_amdgcn_wmma_f32_16x16x4_f32(
                false, a, false, b3, (short)0, acc3, false, false);
        }
        // distance' = ||e||^2 - 2*S (row norm is constant per row -> argmin safe)
#pragma unroll
        for (int t = 0; t < 4; ++t) {
            v8f accv = (t == 0) ? acc0 : (t == 1) ? acc1 : (t == 2) ? acc2 : acc3;
            float c    = cn[ng + t * 16 + m];
            int   code = ng + t * 16 + m;
#pragma unroll
            for (int v = 0; v < 8; ++v) {
                float d = c - 2.0f * accv[v];
                if (d < best[v]) { best[v] = d; bidx[v] = code; } // '<': lowest idx
            }
        }
    }

    // C layout: VGPR v -> row r0+v (lanes 0-15) / r0+v+8 (lanes 16-31); col = lane%16.
    // Reduce min(+lowest index tie-break) across each 16-lane half.
#pragma unroll
    for (int v = 0; v < 8; ++v) {
        float val = best[v];
        int   id  = bidx[v];
#pragma unroll
        for (int off = 8; off >= 1; off >>= 1) {
            float ov = __shfl_xor(val, off, 32);
            int   oi = __shfl_xor(id,  off, 32);
            if (ov < val || (ov == val && oi < id)) { val = ov; id = oi; }
        }
        if (m == 0) {
            int r = r0 + v + 8 * hl;
            idx_out_f[r] = (float)id;
            idx_out_i[r] = id;
        }
    }
}

// ---------------- kernel 3: quantized output + loss partials ----------------
__global__ void quantize_loss_kernel(const float* __restrict__ x,
                                     const float* __restrict__ weight,
                                     const int* __restrict__ idx,
                                     float* __restrict__ qout,
                                     float* __restrict__ partial) {
    __shared__ float red[256];
    size_t tid = (size_t)blockIdx.x * 256 + threadIdx.x;  // == x offset [b,d,h,w]
    int hw = (int)(tid & 1023);
    int d  = (int)((tid >> 10) & 255);
    int b  = (int)(tid >> 18);
    int n  = (b << 10) | hw;                 // flat row index
    int e  = idx[n];
    float q  = weight[e * EMBED_DIM + d];
    float xv = x[tid];
    qout[tid] = q;                           // straight-through value == quantized
    float df = q - xv;
    red[threadIdx.x] = df * df;
    __syncthreads();
    for (int s = 128; s > 0; s >>= 1) {
        if (threadIdx.x < s) red[threadIdx.x] += red[threadIdx.x + s];
        __syncthreads();
    }
    if (threadIdx.x == 0) partial[blockIdx.x] = red[0];
}

// ---------------- kernel 4: deterministic dense "scatter" ----------------
// one block per code: scan idx (L2-resident) in fixed order, sum matching rows.
__global__ void embed_sum_kernel(const float* __restrict__ x,
                                 const int* __restrict__ idx,
                                 float* __restrict__ embed_sum_out,  // d_out embed_avg slot
                                 float* __restrict__ count_out) {    // d_out N slot
    int e = blockIdx.x, d = threadIdx.x;
    float sum = 0.f;
    int   cnt = 0;
#pragma unroll 4
    for (int n = 0; n < M_ROWS; ++n) {
        if (idx[n] == e) {
            int b = n >> 10, hw = n & 1023;
            sum += x[(size_t)b * X_BSTRIDE + (size_t)d * HW + hw];
            cnt++;
        }
    }
    embed_sum_out[(size_t)e * EMBED_DIM + d] = sum;
    if (d == 0) count_out[e] = (float)cnt;
}

// ---------------- kernel 5a: EMA counts + cluster sizes ----------------
__global__ void ema_n_kernel(const float* __restrict__ N_old,
                             float* __restrict__ N_out,      // in: counts, out: N_new
                             float* __restrict__ cluster) {
    __shared__ float red[1024];
    int t = threadIdx.x;
    float cntv = N_out[t];
    float Nn = 0.99f * N_old[t] + 0.01f * cntv;
    red[t] = Nn;
    __syncthreads();
    for (int s = 512; s > 0; s >>= 1) {
        if (t < s) red[t] += red[t + s];
        __syncthreads();
    }
    float nsum = red[0];
    float cs = (Nn + 1e-5f) / (nsum + N_EMBED * 1e-5f) * nsum;
    N_out[t]   = Nn;
    cluster[t] = cs;
}

// ---------------- kernel 5b: EMA embed_avg + new weights ----------------
__global__ void ema_w_kernel(const float* __restrict__ embed_avg_old,
                             const float* __restrict__ cluster,
                             float* __restrict__ ea_out,   // in: embed_sum, out: embed_avg_new
                             float* __restrict__ w_out) {
    int e = blockIdx.x, d = threadIdx.x;
    size_t o = (size_t)e * EMBED_DIM + d;
    float s  = ea_out[o];
    float ea = 0.99f * embed_avg_old[o] + 0.01f * s;
    ea_out[o] = ea;
    w_out[o]  = ea / cluster[e];
}

// ---------------- kernel 6: final loss reduction ----------------
__global__ void loss_reduce_kernel(const float* __restrict__ partial,
                                   float* __restrict__ loss) {
    __shared__ float red[1024];
    int t = threadIdx.x;
    float s = 0.f;
    for (int i = 0; i < 32; ++i) s += partial[t + i * 1024];  // fixed order
    red[t] = s;
    __syncthreads();
    for (int st = 512; st > 0; st >>= 1) {
        if (t < st) red[t] += red[t + st];
        __syncthreads();
    }
    if (t == 0) loss[0] = 0.25f * red[0] / (float)X_TOTAL;
}

extern "C" void kernel_launch(void* const* d_in, const int* in_sizes, int n_in,
                              void* d_out, int out_size, void* d_ws, size_t ws_size,
                              hipStream_t stream) {
    const float* x             = (const float*)d_in[0];  // [32,256,32,32]
    const float* weight        = (const float*)d_in[1];  // [1024,256]
    const float* N_old         = (const float*)d_in[2];  // [1024]
    const float* embed_avg_old = (const float*)d_in[3];  // [1024,256]

    float* out       = (float*)d_out;
    float* q_out     = out;                       // 8388608 quantized_st
    float* idx_out_f = q_out + X_TOTAL;           // 32768  embed_idx (as float)
    float* loss_out  = idx_out_f + M_ROWS;        // 1      latent_loss
    float* N_out     = loss_out + 1;              // 1024   N_new (counts temp)
    float* ea_out    = N_out + N_EMBED;           // 262144 embed_avg_new (sum temp)
    float* w_out     = ea_out + N_EMBED * EMBED_DIM; // 262144 weight_new

    char*  ws         = (char*)d_ws;
    float* ws_cn      = (float*)ws;                                    // 1024
    int*   ws_idx     = (int*)(ws + 4 * (N_EMBED));                    // 32768
    float* ws_partial = (float*)(ws + 4 * (N_EMBED + M_ROWS));         // 32768
    float* ws_cluster = (float*)(ws + 4 * (N_EMBED + 2 * M_ROWS));     // 1024

    code_norm_kernel   <<<N_EMBED,        256, 0, stream>>>(weight, ws_cn);
    argmin_wmma_kernel <<<M_ROWS / 128,   256, 0, stream>>>(x, weight, ws_cn,
                                                            idx_out_f, ws_idx);
    quantize_loss_kernel<<<X_TOTAL / 256, 256, 0, stream>>>(x, weight, ws_idx,
                                                            q_out, ws_partial);
    embed_sum_kernel   <<<N_EMBED,        256, 0, stream>>>(x, ws_idx, ea_out, N_out);
    ema_n_kernel       <<<1,             1024, 0, stream>>>(N_old, N_out, ws_cluster);
    ema_w_kernel       <<<1024,           256, 0, stream>>>(embed_avg_old, ws_cluster,
                                                            ea_out, w_out);
    loss_reduce_kernel <<<1,             1024, 0, stream>>>(ws_partial, loss_out);
}